// RydbergHamiltonian_15693810500007
// MI455X (gfx1250) — compile-verified
//
#include <hip/hip_runtime.h>

// Rydberg Hamiltonian apply: out[k] = diag(k)*state[k] + sum_p 0.5*rabi(p)*state[k^(1<<p)]
// N = 22 qubits, DIM = 2^22. Qubit i (reference numbering, 0 = MSB) lives at
// bit position p = 21 - i of the basis index.
//
// MI455X plan: state (16MB) is L2-resident (192MB L2) -> all 22 flip gathers are
// L2 hits; HBM traffic ~32MB => ~1.4us floor at 23.3 TB/s. Per block: 4096-element
// tile staged to LDS with CDNA5 global_load_async_to_lds_b128 (ASYNCcnt path).
// Flips: bits 0-1 in registers, bits 2-11 via ds_load_b128, bits 12-21 via
// coalesced global b128 (L2 hits). Diagonal quadratic form factored so the
// per-element VALU cost is ~18 FMA.

#define NQ   22
#define TILE 4096
#define TPB  256

// ---------------- main kernel (defined FIRST so disasm/histogram shows it) ----
__global__ __launch_bounds__(TPB) void ryd_main(const float* __restrict__ state,
                                                const float* __restrict__ W,
                                                const float* __restrict__ detC,
                                                const float* __restrict__ rc,
                                                float* __restrict__ out) {
    __shared__ float4 ldsF4[TILE / 4];

    const unsigned tid      = threadIdx.x;
    const unsigned tileBase = blockIdx.x * (unsigned)TILE;  // sets bits 12..21 of k

    // LDS byte address of the tile, taken from the pointer itself:
    //  - local->flat addrspacecast keeps the LDS byte offset in the low 32 bits
    //  - the ptrtoint CAPTURES ldsF4, so the asm "memory" clobbers below are
    //    known to alias it (otherwise the LDS reads fold to poison/NaN).
    const unsigned ldsBase = (unsigned)(size_t)(void*)ldsF4;

    // ---- stage 16KB tile into LDS via CDNA5 async DMA (overlaps diag precompute)
    {
        const float* gsrc = state + tileBase;   // uniform -> SGPR pair (saddr)
        unsigned byteOff = tid * 16u;
        #pragma unroll
        for (int i = 0; i < 4; ++i) {
            unsigned goff = byteOff + (unsigned)i * 4096u;
            unsigned loff = ldsBase + goff;
            asm volatile("global_load_async_to_lds_b128 %0, %1, %2"
                         :
                         : "v"(loff), "v"(goff), "s"(gsrc)
                         : "memory");
        }
    }

    // ---- per-thread diag precompute (overlaps the async DMA) ----
    // k = tileBase + v*1024 + tid*4 + r : fixed bits {2..9, 12..21},
    // variable bits {10,11} (v) and {0,1} (r).
    const unsigned Kf = tileBase + tid * 4u;
    float b[NQ];
    #pragma unroll
    for (int p = 0; p < NQ; ++p) b[p] = (float)((Kf >> p) & 1u);

    float dFix = 0.0f;
    float g0 = detC[0], g1 = detC[1], g10 = detC[10], g11 = detC[11];
    #pragma unroll
    for (int p = 0; p < NQ; ++p) {
        if (p == 0 || p == 1 || p == 10 || p == 11) continue;
        float inner = detC[p];
        #pragma unroll
        for (int q = p + 1; q < NQ; ++q) {
            if (q == 10 || q == 11) continue;
            inner = fmaf(b[q], W[p * NQ + q], inner);
        }
        dFix = fmaf(b[p], inner, dFix);
        g0   = fmaf(b[p], W[p * NQ + 0],  g0);
        g1   = fmaf(b[p], W[p * NQ + 1],  g1);
        g10  = fmaf(b[p], W[p * NQ + 10], g10);
        g11  = fmaf(b[p], W[p * NQ + 11], g11);
    }
    const float W01    = W[0 * NQ + 1];
    const float W10_11 = W[10 * NQ + 11];
    const float W10_0  = W[10 * NQ + 0], W11_0 = W[11 * NQ + 0];
    const float W10_1  = W[10 * NQ + 1], W11_1 = W[11 * NQ + 1];

    float rcv[NQ];  // uniform addresses -> s_load into SGPRs
    #pragma unroll
    for (int p = 0; p < NQ; ++p) rcv[p] = rc[p];

    // ---- wait for this wave's async staging, then block barrier ----
    asm volatile("s_wait_asynccnt 0" ::: "memory");
    __syncthreads();

    const float4* stF4  = (const float4*)state;
    float4*       outF4 = (float4*)out;

    #pragma unroll
    for (int v = 0; v < 4; ++v) {
        const float bv10 = (float)(v & 1);
        const float bv11 = (float)((v >> 1) & 1);
        float dv = dFix;
        dv = fmaf(bv10, g10, dv);
        dv = fmaf(bv11, g11, dv);
        dv = fmaf(bv10 * bv11, W10_11, dv);
        float h0 = g0; h0 = fmaf(bv10, W10_0, h0); h0 = fmaf(bv11, W11_0, h0);
        float h1 = g1; h1 = fmaf(bv10, W10_1, h1); h1 = fmaf(bv11, W11_1, h1);

        const unsigned f  = tid + (unsigned)v * 256u;     // float4 index in tile
        const unsigned Fg = (tileBase >> 2) + f;          // global float4 index

        float4 s = ldsF4[f];                              // ds_load_b128
        float4 o;
        o.x = dv * s.x;
        o.y = (dv + h0) * s.y;
        o.z = (dv + h1) * s.z;
        o.w = (dv + h0 + h1 + W01) * s.w;

        // bit 0 flip (x<->y, z<->w) and bit 1 flip (x<->z, y<->w): registers
        o.x = fmaf(rcv[0], s.y, o.x);  o.y = fmaf(rcv[0], s.x, o.y);
        o.z = fmaf(rcv[0], s.w, o.z);  o.w = fmaf(rcv[0], s.z, o.w);
        o.x = fmaf(rcv[1], s.z, o.x);  o.y = fmaf(rcv[1], s.w, o.y);
        o.z = fmaf(rcv[1], s.x, o.z);  o.w = fmaf(rcv[1], s.y, o.w);

        // bits 2..11: partner float4 inside the LDS tile (ds_load_b128)
        #pragma unroll
        for (int p = 2; p <= 11; ++p) {
            float4 t = ldsF4[f ^ (1u << (p - 2))];
            float  c = rcv[p];
            o.x = fmaf(c, t.x, o.x);  o.y = fmaf(c, t.y, o.y);
            o.z = fmaf(c, t.z, o.z);  o.w = fmaf(c, t.w, o.w);
        }
        // bits 12..21: partner float4 in another tile (coalesced, L2-resident)
        #pragma unroll
        for (int p = 12; p <= 21; ++p) {
            float4 t = stF4[Fg ^ (1u << (p - 2))];
            float  c = rcv[p];
            o.x = fmaf(c, t.x, o.x);  o.y = fmaf(c, t.y, o.y);
            o.z = fmaf(c, t.z, o.z);  o.w = fmaf(c, t.w, o.w);
        }
        outF4[Fg] = o;
    }
}

// ---------------- setup: bit-position-indexed coefficient tables --------------
// W[p*22+q] (p!=q): U between qubits (21-p),(21-q), symmetric. W[p][p] = 0.
// detC[p] = -detune[21-p].  rc[p] = 0.5*rabi[21-p].
__global__ void ryd_setup(const float* __restrict__ rabi,
                          const float* __restrict__ detune,
                          const float* __restrict__ U,
                          float* __restrict__ W,
                          float* __restrict__ detC,
                          float* __restrict__ rc) {
    int t = threadIdx.x;
    if (t < NQ) {
        detC[t] = -detune[NQ - 1 - t];
        rc[t]   = 0.5f * rabi[NQ - 1 - t];
    }
    for (int e = t; e < NQ * NQ; e += blockDim.x) {
        int p = e / NQ, q = e % NQ;
        float val = 0.0f;
        if (p != q) {
            int a = NQ - 1 - p, bq = NQ - 1 - q;
            int i = a < bq ? a : bq;
            int j = a < bq ? bq : a;
            // np.triu_indices(22, k=1) row-major flat index
            int tt = i * NQ - (i * (i + 1)) / 2 + (j - i - 1);
            val = U[tt];
        }
        W[e] = val;
    }
}

extern "C" void kernel_launch(void* const* d_in, const int* in_sizes, int n_in,
                              void* d_out, int out_size, void* d_ws, size_t ws_size,
                              hipStream_t stream) {
    const float* state  = (const float*)d_in[0];
    const float* rabi   = (const float*)d_in[1];
    const float* detune = (const float*)d_in[2];
    const float* U      = (const float*)d_in[3];
    float* out = (float*)d_out;

    float* W    = (float*)d_ws;      // 484 floats
    float* detC = W + NQ * NQ;       // 22 floats
    float* rc   = detC + NQ;         // 22 floats

    ryd_setup<<<1, TPB, 0, stream>>>(rabi, detune, U, W, detC, rc);

    const int dim     = out_size;    // 2^22
    const int nblocks = dim / TILE;  // 1024
    ryd_main<<<nblocks, TPB, 0, stream>>>(state, W, detC, rc, out);
}